// IBF_Trainable_K_88545045774890
// MI455X (gfx1250) — compile-verified
//
#include <hip/hip_runtime.h>
#include <hip/hip_bf16.h>
#include <cstddef>

// ---------------------------------------------------------------------------
// Wiener inverse filter via matmul-DFT on CDNA5 WMMA (fp32, 16x16x4).
//   fft2(A)  = F * A * F          (F symmetric DFT matrix, 512x512)
//   ifft2(X) = I * X * I          (I = (1/512) * conj-DFT matrix per dim)
// Six 512^3 complex GEMM passes + one elementwise combine.
// Complex MAC realized as three accumulators so the K-loop body contains
// ONLY loads + v_wmma (no per-iteration VALU; f32 WMMA doesn't co-execute).
// ---------------------------------------------------------------------------

#define Bn 128
#define Nd 512            // H == W == 512
#define FELEMS (Nd * Nd)  // 262144 floats per plane

typedef float v2f __attribute__((ext_vector_type(2)));
typedef float v8f __attribute__((ext_vector_type(8)));

__device__ __forceinline__ v8f wmma4(v2f a, v2f b, v8f c) {
    // D = A(16x4) * B(4x16) + C(16x16), all fp32
    return __builtin_amdgcn_wmma_f32_16x16x4_f32(
        /*neg_a=*/false, a, /*neg_b=*/false, b,
        /*c_mod=*/(short)0, c, /*reuse_a=*/false, /*reuse_b=*/false);
}

// A fragment: 16x4 f32. lanes 0-15: K=k0+0/1 ; lanes 16-31: K=k0+2/3
__device__ __forceinline__ v2f load_a(const float* A, int row0, int k0, int lane) {
    const int m  = lane & 15;
    const int ka = (lane >> 4) << 1;
    return *(const v2f*)(A + (size_t)(row0 + m) * Nd + (k0 + ka));
}

// B fragment: 4x16 f32. lane gives column n; VGPR pair gives rows k0+{0,1}/{2,3}
__device__ __forceinline__ v2f load_b(const float* Bp, int k0, int col0, int lane) {
    const int n  = lane & 15;
    const int kb = (lane >> 4) << 1;
    const float* p = Bp + (size_t)(k0 + kb) * Nd + (col0 + n);
    v2f r; r.x = p[0]; r.y = p[Nd];
    return r;
}

// One 16x16 output tile per wave; 512x512 GEMM, K stepped by 4.
// ACPLX: A has an imaginary plane. BCPLX: B has one. OIMAG: write imag output.
template <bool ACPLX, bool BCPLX, bool OIMAG>
__global__ __launch_bounds__(128) void cgemm512_wmma(
    const float* __restrict__ Are, const float* __restrict__ Aim, size_t strideA,
    const float* __restrict__ Bre, const float* __restrict__ Bim, size_t strideB,
    float* __restrict__ Cre, float* __restrict__ Cim, size_t strideC) {

    const int lane  = threadIdx.x & 31;
    const int wave  = threadIdx.x >> 5;
    const int col0  = (blockIdx.x * 4 + wave) * 16;
    const int row0  = blockIdx.y * 16;
    const size_t im = blockIdx.z;

    const float* Ar = Are + im * strideA;
    const float* Ai = ACPLX ? (Aim + im * strideA) : nullptr;
    const float* Br = Bre + im * strideB;
    const float* Bi = BCPLX ? (Bim + im * strideB) : nullptr;

    v8f accRe = {};   // sum Ar*Br
    v8f accMM = {};   // sum Ai*Bi  (subtracted once at the end)
    v8f accIm = {};   // sum Ar*Bi + Ai*Br

#pragma unroll 4
    for (int k0 = 0; k0 < Nd; k0 += 4) {
        v2f ar = load_a(Ar, row0, k0, lane);
        v2f br = load_b(Br, k0, col0, lane);
        accRe = wmma4(ar, br, accRe);                 // + Ar*Br

        if (BCPLX) {
            v2f bi = load_b(Bi, k0, col0, lane);
            if (ACPLX) {
                v2f ai = load_a(Ai, row0, k0, lane);
                accMM = wmma4(ai, bi, accMM);         // + Ai*Bi (sign at end)
                if (OIMAG) {
                    accIm = wmma4(ar, bi, accIm);     // + Ar*Bi
                    accIm = wmma4(ai, br, accIm);     // + Ai*Br
                }
            } else if (OIMAG) {
                accIm = wmma4(ar, bi, accIm);         // real A: Im = A*Bi
            }
        }
    }

    // C/D layout: VGPR j -> M = row0 + j + 8*(lane>=16), N = col0 + (lane&15)
    const int g = (lane >> 4) << 3;
    const int n = lane & 15;
    float* Cr = Cre + im * strideC;
    float* Ci = OIMAG ? (Cim + im * strideC) : nullptr;
#pragma unroll
    for (int j = 0; j < 8; ++j) {
        const size_t idx = (size_t)(row0 + g + j) * Nd + (col0 + n);
        Cr[idx] = (ACPLX && BCPLX) ? (accRe[j] - accMM[j]) : accRe[j];
        if (OIMAG) Ci[idx] = accIm[j];
    }
}

// Build forward DFT (Fc + i*Fs, Fs = -sin) and scaled inverse DFT
// (Ic + i*Is, each carrying a 1/512 factor so ifft2 total scale = 1/512^2).
__global__ __launch_bounds__(256) void build_dft(float* __restrict__ Fc,
                                                 float* __restrict__ Fs,
                                                 float* __restrict__ Ic,
                                                 float* __restrict__ Is) {
    const int idx = blockIdx.x * blockDim.x + threadIdx.x;
    if (idx >= FELEMS) return;
    const int nrow = idx >> 9;
    const int kcol = idx & (Nd - 1);
    const int t    = (nrow * kcol) & (Nd - 1);     // exact (nk mod 512)
    const float ang = (float)t * (6.2831853071795864769f / (float)Nd);
    float s, c;
    __sincosf(ang, &s, &c);
    Fc[idx] = c;
    Fs[idx] = -s;
    Ic[idx] = c * (1.0f / (float)Nd);
    Is[idx] = s * (1.0f / (float)Nd);
}

// X = conj(Fk)*Fy / (|Fk|^2 + relu(K_w) + 1e-5)
__global__ __launch_bounds__(256) void wiener_combine(
    const float* __restrict__ FyRe, const float* __restrict__ FyIm,
    const float* __restrict__ FkRe, const float* __restrict__ FkIm,
    const float* __restrict__ Kw,
    float* __restrict__ Xre, float* __restrict__ Xim, size_t total) {
    const size_t p = (size_t)blockIdx.x * blockDim.x + threadIdx.x;
    if (p >= total) return;
    const size_t hw = p & (size_t)(FELEMS - 1);
    const float yr = FyRe[p], yi = FyIm[p];
    const float kr = FkRe[p], ki = FkIm[p];
    const float reg = fmaxf(Kw[hw], 0.0f) + 1e-5f;
    const float inv = 1.0f / (kr * kr + ki * ki + reg);
    Xre[p] = (kr * yr + ki * yi) * inv;
    Xim[p] = (kr * yi - ki * yr) * inv;
}

extern "C" void kernel_launch(void* const* d_in, const int* in_sizes, int n_in,
                              void* d_out, int out_size, void* d_ws, size_t ws_size,
                              hipStream_t stream) {
    (void)in_sizes; (void)n_in; (void)out_size; (void)ws_size;

    const float* y  = (const float*)d_in[0];   // (128,512,512)
    const float* k  = (const float*)d_in[1];   // (128,512,512)
    const float* Kw = (const float*)d_in[2];   // (1,512,512)
    float* out = (float*)d_out;

    const size_t P = (size_t)FELEMS;          // plane size (per image)
    const size_t N = (size_t)Bn * P;          // full batch plane

    float* ws = (float*)d_ws;
    float* Fc = ws;                 // forward DFT real
    float* Fs = ws + 1 * P;         // forward DFT imag (= -sin)
    float* Ic = ws + 2 * P;         // inverse DFT real (scaled 1/512)
    float* Is = ws + 3 * P;         // inverse DFT imag (scaled 1/512)
    float* T1re = ws + 4 * P;             // batch scratch (re)
    float* T1im = T1re + N;               // batch scratch (im)
    float* FyRe = T1im + N;
    float* FyIm = FyRe + N;
    float* FkRe = FyIm + N;
    float* FkIm = FkRe + N;

    // 0) DFT matrices (1 MB each; stay L2-resident for all GEMM passes)
    build_dft<<<FELEMS / 256, 256, 0, stream>>>(Fc, Fs, Ic, Is);

    const dim3 gg(Nd / 16 / 4, Nd / 16, Bn);   // 8 x 32 x 128 blocks
    const dim3 gb(128);                        // 4 waves, 1 tile each

    // 1) row-FFT of y: T1 = y * F           (real A, complex B)
    cgemm512_wmma<false, true, true><<<gg, gb, 0, stream>>>(
        y, nullptr, P, Fc, Fs, 0, T1re, T1im, P);
    // 2) col-FFT of y: Fy = F * T1          (complex * complex)
    cgemm512_wmma<true, true, true><<<gg, gb, 0, stream>>>(
        Fc, Fs, 0, T1re, T1im, P, FyRe, FyIm, P);
    // 3) row-FFT of k
    cgemm512_wmma<false, true, true><<<gg, gb, 0, stream>>>(
        k, nullptr, P, Fc, Fs, 0, T1re, T1im, P);
    // 4) col-FFT of k
    cgemm512_wmma<true, true, true><<<gg, gb, 0, stream>>>(
        Fc, Fs, 0, T1re, T1im, P, FkRe, FkIm, P);

    // 5) Wiener combine -> X (reuse T1)
    wiener_combine<<<(unsigned)((N + 255) / 256), 256, 0, stream>>>(
        FyRe, FyIm, FkRe, FkIm, Kw, T1re, T1im, N);

    // 6) inverse row pass: T2 = X * I       (reuse Fy buffers as T2)
    cgemm512_wmma<true, true, true><<<gg, gb, 0, stream>>>(
        T1re, T1im, P, Ic, Is, 0, FyRe, FyIm, P);
    // 7) inverse col pass, real part only: out = Re(I * T2)
    cgemm512_wmma<true, true, false><<<gg, gb, 0, stream>>>(
        Ic, Is, 0, FyRe, FyIm, P, out, nullptr, P);
}